// VLAD_47021301957418
// MI455X (gfx1250) — compile-verified
//
#include <hip/hip_runtime.h>

typedef __attribute__((ext_vector_type(2))) float v2f;
typedef __attribute__((ext_vector_type(4))) float v4f;
typedef __attribute__((ext_vector_type(8))) float v8f;

#define NPTS 512
#define DIM  1024
#define KCL  50
#define KP   64
#define NIT  10
#define QD   256   // dims per accumulation pass (LDS sums = 50*256*4 = 50 KB)
#define NQ   4

__global__ __launch_bounds__(512) void vlad_kernel(const float* __restrict__ feat,
                                                   float* __restrict__ out)
{
    __shared__ float sums[KCL * QD];
    __shared__ float cnorm[KP];
    __shared__ int   assign[NPTS];
    __shared__ int   cnt[KP];

    const int b    = blockIdx.x;
    const int tid  = threadIdx.x;
    const int wave = tid >> 5;
    const int lane = tid & 31;
    const int half = lane >> 4;
    const int l15  = lane & 15;

    const float* F = feat + (size_t)b * NPTS * DIM;
    float*       C = out  + (size_t)b * KCL  * DIM;  // centers live here; final VLAD overwrites

    const v8f vzero8 = {0.f,0.f,0.f,0.f,0.f,0.f,0.f,0.f};

    // ---- centers0 = feat[:K] ----
    for (int i = tid; i < KCL * DIM; i += 512) C[i] = F[i];
    __threadfence();
    __syncthreads();

    const int m0 = wave * 32;

    // Hoisted per-lane base pointers for the WMMA operand streams.
    // Dim regrouping: macro-chunk of 8 dims; half-wave h owns dims kd+4h..kd+4h+3.
    // WMMA #1 uses .xy (K = {0,1} of each half), WMMA #2 uses .zw — together they
    // cover all 8 dims exactly once with identical A/B mapping.
    const int dimOfs = 4 * half;
    const float* a0p = F + (size_t)(m0 +      l15) * DIM + dimOfs;
    const float* a1p = F + (size_t)(m0 + 16 + l15) * DIM + dimOfs;
    int browIdx[4];
    #pragma unroll
    for (int t = 0; t < 4; ++t) {
        int kc = t * 16 + l15;
        browIdx[t] = (kc < KCL) ? kc : (KCL - 1);   // clamp; padded k poisoned via cnorm
    }

    for (int it = 0; it <= NIT; ++it) {
        // ---- cluster squared norms; padded clusters poisoned so argmin skips them ----
        {
            int kc = wave * 4 + (lane >> 3);
            int dl = lane & 7;
            float s = 0.0f;
            if (kc < KCL) {
                const float* cp = C + (size_t)kc * DIM;
                for (int d = dl; d < DIM; d += 8) { float c = cp[d]; s += c * c; }
            }
            for (int m = 4; m >= 1; m >>= 1) s += __shfl_xor(s, m, 32);
            if (dl == 0) cnorm[kc] = (kc < KCL) ? s : 1.0e30f;
        }
        __syncthreads();

        // ---- dot = F * C^T via V_WMMA_F32_16X16X4_F32; wave owns 32 points x 64 clusters ----
        v8f acc[2][4];
        #pragma unroll
        for (int i = 0; i < 2; ++i)
            #pragma unroll
            for (int t = 0; t < 4; ++t) acc[i][t] = vzero8;

        const float* bp[4];
        #pragma unroll
        for (int t = 0; t < 4; ++t) bp[t] = C + (size_t)browIdx[t] * DIM + dimOfs;

        for (int kd = 0; kd < DIM; kd += 8) {
            v4f a0 = *(const v4f*)(a0p + kd);
            v4f a1 = *(const v4f*)(a1p + kd);
            v4f bt[4];
            #pragma unroll
            for (int t = 0; t < 4; ++t) bt[t] = *(const v4f*)(bp[t] + kd);
            #pragma unroll
            for (int t = 0; t < 4; ++t) {
                acc[0][t] = __builtin_amdgcn_wmma_f32_16x16x4_f32(
                    false, a0.xy, false, bt[t].xy, (short)0, acc[0][t], false, false);
                acc[1][t] = __builtin_amdgcn_wmma_f32_16x16x4_f32(
                    false, a1.xy, false, bt[t].xy, (short)0, acc[1][t], false, false);
            }
            #pragma unroll
            for (int t = 0; t < 4; ++t) {
                acc[0][t] = __builtin_amdgcn_wmma_f32_16x16x4_f32(
                    false, a0.zw, false, bt[t].zw, (short)0, acc[0][t], false, false);
                acc[1][t] = __builtin_amdgcn_wmma_f32_16x16x4_f32(
                    false, a1.zw, false, bt[t].zw, (short)0, acc[1][t], false, false);
            }
        }

        // ---- argmin_k (cnorm[k] - 2*dot) per point, first-index tie-break ----
        float cn[4];
        #pragma unroll
        for (int t = 0; t < 4; ++t) cn[t] = cnorm[t * 16 + l15];

        #pragma unroll
        for (int i = 0; i < 2; ++i) {
            #pragma unroll
            for (int r = 0; r < 8; ++r) {
                float bv = 3.0e38f;
                int   bi = 0;
                #pragma unroll
                for (int t = 0; t < 4; ++t) {
                    float v  = cn[t] - 2.0f * acc[i][t][r];
                    int   kc = t * 16 + l15;
                    if (v < bv) { bv = v; bi = kc; }
                }
                #pragma unroll
                for (int m = 8; m >= 1; m >>= 1) {   // xor stays within 16-lane halves
                    float ov = __shfl_xor(bv, m, 32);
                    int   oi = __shfl_xor(bi, m, 32);
                    if (ov < bv || (ov == bv && oi < bi)) { bv = ov; bi = oi; }
                }
                if (l15 == 0) assign[m0 + 16 * i + 8 * half + r] = bi;
            }
        }
        __syncthreads();

        // ---- counts ----
        if (tid < KP) cnt[tid] = 0;
        __syncthreads();
        atomicAdd(&cnt[assign[tid]], 1);
        __syncthreads();

        // ---- one-hot scatter-add sums; update centers / emit VLAD; dims in 4 passes ----
        const int dq = tid & (QD - 1);
        const int po = tid >> 8;                 // two points in flight across the block
        for (int q = 0; q < NQ; ++q) {
            for (int i = tid; i < KCL * QD; i += 512) sums[i] = 0.0f;
            __syncthreads();
            const int gd = q * QD + dq;
            for (int n = 0; n < NPTS; n += 2) {
                int p = n + po;
                int a = assign[p];
                atomicAdd(&sums[a * QD + dq], F[(size_t)p * DIM + gd]);
            }
            __syncthreads();
            for (int i = tid; i < KCL * QD; i += 512) {
                int k = i >> 8;                  // QD == 256
                int d = (i & (QD - 1)) + q * QD;
                float* cp   = C + (size_t)k * DIM + d;
                float  cold = *cp;
                float  cv   = (float)cnt[k];
                if (it < NIT)
                    *cp = (cnt[k] > 0) ? (sums[i] / cv) : cold;   // Lloyd update
                else
                    *cp = sums[i] - cv * cold;                    // VLAD residual
            }
            __syncthreads();
        }
        __threadfence();
        __syncthreads();
    }
}

extern "C" void kernel_launch(void* const* d_in, const int* in_sizes, int n_in,
                              void* d_out, int out_size, void* d_ws, size_t ws_size,
                              hipStream_t stream) {
    const float* x  = (const float*)d_in[0];
    float* out      = (float*)d_out;
    int batches     = in_sizes[0] / (NPTS * DIM);   // 64 for the reference shapes
    vlad_kernel<<<dim3(batches), dim3(512), 0, stream>>>(x, out);
}